// MultiHeadPool_45226005627089
// MI455X (gfx1250) — compile-verified
//
#include <hip/hip_runtime.h>

typedef __attribute__((ext_vector_type(2))) float v2f;
typedef __attribute__((ext_vector_type(8))) float v8f;

// Exact parameter element type from the compiler diagnostic:
// '__attribute__((__vector_size__(4 * sizeof(int)))) int'
typedef int v4i __attribute__((vector_size(4 * sizeof(int))));
typedef __attribute__((address_space(1))) v4i* gv4i_ptr;  // global src
typedef __attribute__((address_space(3))) v4i* lv4i_ptr;  // LDS dst

// LDS row strides (in floats), padded so fragment loads spread across banks:
// bank(row,col) = (stride*row + col) % 64 ; stride%64 == 4 -> conflict-free
#define TS 132   // tile rows: 128 + 4
#define QS 132   // query rows: 128 + 4
#define WS 68    // weight rows: 64 + 4

#if defined(__has_builtin)
#  if __has_builtin(__builtin_amdgcn_global_load_async_to_lds_b128)
#    define USE_ASYNC_LDS 1
#  endif
#  if __has_builtin(__builtin_amdgcn_s_wait_asynccnt)
#    define HAVE_WAIT_ASYNC_BUILTIN 1
#  endif
#endif

__device__ __forceinline__ void wait_asynccnt0() {
#if defined(USE_ASYNC_LDS)
#  if defined(HAVE_WAIT_ASYNC_BUILTIN)
    __builtin_amdgcn_s_wait_asynccnt(0);
#  else
    asm volatile("s_wait_asynccnt 0" ::: "memory");
#  endif
#endif
}

__global__ __launch_bounds__(128) void MultiHeadPool_kernel(
    const float* __restrict__ others,   // (B=8, N=64, T=512, D=128)
    const float* __restrict__ queries,  // (H=7, D=128)
    float* __restrict__ out)            // (B=8, T=512, H=7, D=128)
{
    constexpr int Nn = 64, Tn = 512, Dn = 128, Hn = 7;
    constexpr float scale = 0.08838834764831845f; // 128^-0.5

    __shared__ float tile[64 * TS];  // others[b,:,t,:]  (N x D)
    __shared__ float Qs[16 * QS];    // queries padded to 16 rows, pre-scaled
    __shared__ float Ws[16 * WS];    // logits -> softmax weights (16 x N)

    const int t    = blockIdx.x;
    const int b    = blockIdx.y;
    const int tid  = threadIdx.x;
    const int wave = tid >> 5;
    const int lane = tid & 31;
    const int hl   = lane & 15;   // half-lane
    const int hi   = lane >> 4;   // which half of the wave

    // ---- Stage 1: others[b,:,t,:] -> LDS, one pass (32 KB) ----
    // Each row n is 128 contiguous floats; rows strided by T*D in memory.
    // CDNA5 path: async global->LDS (ASYNCcnt), no VGPR round-trip.
#if defined(USE_ASYNC_LDS)
    {
        const int d4 = (lane << 2);
        for (int n0 = 0; n0 < Nn; n0 += 4) {
            const int n = n0 + wave;
            const float* gp =
                others + (((size_t)b * Nn + n) * Tn + t) * Dn + d4;
            __builtin_amdgcn_global_load_async_to_lds_b128(
                (gv4i_ptr)(void*)gp,
                (lv4i_ptr)&tile[n * TS + d4],
                /*offset=*/0, /*cpol=*/0);
        }
    }
#else
    {
        const int d4 = (lane << 2);
        for (int n0 = 0; n0 < Nn; n0 += 4) {
            const int n = n0 + wave;
            const float4 v = *(const float4*)(
                others + (((size_t)b * Nn + n) * Tn + t) * Dn + d4);
            *(float4*)(&tile[n * TS + d4]) = v;
        }
    }
#endif

    // ---- Stage 0 (overlapped with async copy): queries -> LDS ----
    // Scaled by D^-1/2, zero-padded rows 7..15.
    for (int idx = tid; idx < 16 * Dn; idx += 128) {
        const int h = idx >> 7, d = idx & 127;
        Qs[h * QS + d] = (h < Hn) ? queries[h * Dn + d] * scale : 0.0f;
    }

    wait_asynccnt0();   // tile copy complete (per-wave)
    __syncthreads();    // visible to whole workgroup

    // ---- Stage 2: logits = Qs (16x128) x tile^T (128x64), f32 WMMA ----
    // Wave w computes the 16x16 tile at columns n_off..n_off+15.
    {
        v8f acc = {0.f, 0.f, 0.f, 0.f, 0.f, 0.f, 0.f, 0.f};
        const int n_off = wave << 4;
        #pragma unroll 4
        for (int k0 = 0; k0 < Dn; k0 += 4) {
            const int col = k0 + (hi << 1);
            // A frag: A[M=hl][K=col..col+1]
            const v2f a = *(const v2f*)(&Qs[hl * QS + col]);
            // B frag: B[K][N] = tile[n][d] -> tile[n_off+hl][col..col+1]
            const v2f bb = *(const v2f*)(&tile[(n_off + hl) * TS + col]);
            acc = __builtin_amdgcn_wmma_f32_16x16x4_f32(
                false, a, false, bb, (short)0, acc, false, false);
        }
        // c[i] = C[M=i+8*hi][N=hl] -> spill logits tile to LDS
        #pragma unroll
        for (int i = 0; i < 8; ++i)
            Ws[(i + (hi << 3)) * WS + n_off + hl] = acc[i];
    }
    __syncthreads();

    // ---- Stage 3: softmax over N per head row (rows 7..15 are padding) ----
    if (tid < 16) {
        float* row = &Ws[tid * WS];
        float m = row[0];
        for (int j = 1; j < Nn; ++j) m = fmaxf(m, row[j]);
        float s = 0.f;
        for (int j = 0; j < Nn; ++j) {
            const float e = __expf(row[j] - m);
            row[j] = e;
            s += e;
        }
        const float inv = 1.0f / s;
        for (int j = 0; j < Nn; ++j) row[j] *= inv;
    }
    __syncthreads();

    // ---- Stage 4: ctx = Ws (16x64) x tile (64x128), f32 WMMA ----
    // Wave w computes 16x32 slab at columns d_off..d_off+31 (two 16x16 tiles).
    {
        v8f acc0 = {0.f, 0.f, 0.f, 0.f, 0.f, 0.f, 0.f, 0.f};
        v8f acc1 = {0.f, 0.f, 0.f, 0.f, 0.f, 0.f, 0.f, 0.f};
        const int d_off = wave << 5;
        #pragma unroll 4
        for (int k0 = 0; k0 < Nn; k0 += 4) {
            const int kr = k0 + (hi << 1);
            // A frag: W[M=hl][K=kr..kr+1] (contiguous in LDS row)
            const v2f a = *(const v2f*)(&Ws[hl * WS + kr]);
            // B frag: B[K=n][N=d] = tile[n][d] -> same column, rows kr, kr+1
            v2f b0, b1;
            b0.x = tile[kr * TS + d_off + hl];
            b0.y = tile[(kr + 1) * TS + d_off + hl];
            b1.x = tile[kr * TS + d_off + 16 + hl];
            b1.y = tile[(kr + 1) * TS + d_off + 16 + hl];
            acc0 = __builtin_amdgcn_wmma_f32_16x16x4_f32(
                false, a, false, b0, (short)0, acc0, false, false);
            acc1 = __builtin_amdgcn_wmma_f32_16x16x4_f32(
                false, a, false, b1, (short)0, acc1, false, false);
        }
        // Store rows h = 0..6 (low half-wave holds rows 0..7), coalesced.
        if (hi == 0) {
            float* obase = out + (((size_t)b * Tn + t) * Hn) * Dn;
            #pragma unroll
            for (int i = 0; i < Hn; ++i) {
                obase[i * Dn + d_off + hl]      = acc0[i];
                obase[i * Dn + d_off + 16 + hl] = acc1[i];
            }
        }
    }
}

extern "C" void kernel_launch(void* const* d_in, const int* in_sizes, int n_in,
                              void* d_out, int out_size, void* d_ws, size_t ws_size,
                              hipStream_t stream) {
    // inputs: d_in[0] = ego (unused, shape-only), d_in[1] = others, d_in[2] = queries
    const float* others  = (const float*)d_in[1];
    const float* queries = (const float*)d_in[2];
    float* out = (float*)d_out;

    dim3 grid(512, 8);  // (T, B)
    dim3 block(128);    // 4 waves
    hipLaunchKernelGGL(MultiHeadPool_kernel, grid, block, 0, stream,
                       others, queries, out);
}